// OutlookAttention_1580547971996
// MI455X (gfx1250) — compile-verified
//
#include <hip/hip_runtime.h>
#include <math.h>
#include <stdint.h>

// ---------------- problem constants ----------------
#define DIM      192
#define HEADS    6
#define HEAD_DIM 32
#define KWIN     3
#define KK       9          // KWIN*KWIN
#define ACOLS    486        // KWIN^4 * HEADS
#define NB       16
#define NH       56
#define NW       56
#define HO       28
#define WO       28
#define NPIX     (NB*NH*NW)   // 50176
#define NPOOL    (NB*HO*WO)   // 12544
#define ASCALE   0.17677669529663687f   // HEAD_DIM^-0.5

typedef float v2f __attribute__((ext_vector_type(2)));
typedef float v8f __attribute__((ext_vector_type(8)));

// =====================================================================
// fp32 WMMA GEMM, register-blocked over N:
//   C[M,N] = A[M,K] * B[K,N] (+ bias[N])
//   - grid = (M/64, ceil(N/(16*NT))), block = 128 (4 waves)
//   - each wave computes a 16 x (16*NT) strip: NT accumulators, one A
//     fragment feeds NT WMMAs per K-step (4x fewer A loads, 4:1 wmma:vmem)
//   - B panel K x (16*NT) staged in LDS; full aligned tiles use
//     GLOBAL_LOAD_ASYNC_TO_LDS_B128 (ASYNCcnt path), ragged tiles use a
//     masked scalar path with zero fill
//   - requires: M % 64 == 0, K % 4 == 0, K <= 192
// Fragment layouts per CDNA5 ISA 7.12.2 (half = lane>>4, m = lane&15):
//   A 16x4 : lane holds A[m][k+2*half], A[m][k+2*half+1]
//   B 4x16 : VGPR0 = rows k / k+2 ; VGPR1 = rows k+1 / k+3
//   C 16x16: VGPR r = row r (lanes0-15) / row r+8 (lanes16-31), col = m
// =====================================================================
template<int NT>
__global__ __launch_bounds__(128)
void gemm_wmma_nt(const float* __restrict__ A, const float* __restrict__ B,
                  const float* __restrict__ bias, float* __restrict__ C,
                  int M, int N, int K)
{
    constexpr int NCOL = 16 * NT;
    __shared__ float Bs[192 * NCOL];          // 48 KB for NT=4
    const int tid   = threadIdx.x;
    const int lane  = tid & 31;
    const int wave  = tid >> 5;
    const int mtile = (blockIdx.x * 4 + wave) * 16;
    const int nbase = blockIdx.y * NCOL;

    // ---- stage B panel (K x NCOL) into LDS ----
    const bool full_aligned = (nbase + NCOL <= N) && ((N & 3) == 0);
    if (full_aligned) {
        // async DMA: 16 bytes/lane straight into LDS, no VGPR round trip
        for (int idx = tid; idx < K * (NCOL / 4); idx += 128) {
            const int k  = idx / (NCOL / 4);
            const int c4 = idx % (NCOL / 4);
            const float* gsrc = B + (size_t)k * N + nbase + c4 * 4;
            const unsigned ldst =
                (unsigned)(unsigned long long)(&Bs[k * NCOL + c4 * 4]);
            asm volatile("global_load_async_to_lds_b128 %0, %1, off"
                         :: "v"(ldst), "v"(gsrc) : "memory");
        }
        asm volatile("s_wait_asynccnt 0x0" ::: "memory");
    } else {
        for (int idx = tid; idx < K * NCOL; idx += 128) {
            const int k   = idx / NCOL;
            const int c   = idx % NCOL;
            const int col = nbase + c;
            Bs[idx] = (col < N) ? B[(size_t)k * N + col] : 0.0f;
        }
    }
    __syncthreads();

    const int half = lane >> 4;               // 0 or 1
    const int m    = lane & 15;

    const v2f* Arow = (const v2f*)(A + (size_t)(mtile + m) * K);

    v8f acc[NT];
#pragma unroll
    for (int nt = 0; nt < NT; ++nt) acc[nt] = (v8f){};

    for (int kk = 0; kk < K; kk += 4) {
        // A fragment: contiguous pair A[m][kk+2*half .. +1] (8-byte load)
        const v2f af = Arow[(kk >> 1) + half];
#pragma unroll
        for (int nt = 0; nt < NT; ++nt) {
            v2f bf;
            bf.x = Bs[(kk + 2 * half)     * NCOL + nt * 16 + m];
            bf.y = Bs[(kk + 2 * half + 1) * NCOL + nt * 16 + m];
            acc[nt] = __builtin_amdgcn_wmma_f32_16x16x4_f32(
                          /*neg_a=*/false, af, /*neg_b=*/false, bf,
                          /*c_mod=*/(short)0, acc[nt],
                          /*reuse_a=*/false, /*reuse_b=*/false);
        }
    }

#pragma unroll
    for (int nt = 0; nt < NT; ++nt) {
        const int col = nbase + nt * 16 + m;
        if (col < N) {
            const float bv = bias ? bias[col] : 0.0f;
#pragma unroll
            for (int r = 0; r < 8; ++r) {
                const int row = mtile + r + 8 * half;
                C[(size_t)row * N + col] = acc[nt][r] + bv;
            }
        }
    }
}

// =====================================================================
// 2x2 average pool:  pooled[b,hn,wn,c] = mean of x over the 2x2 block
// =====================================================================
__global__ __launch_bounds__(256)
void pool_kernel(const float* __restrict__ x, float* __restrict__ pooled)
{
    const int idx = blockIdx.x * 256 + threadIdx.x;
    if (idx >= NPOOL * DIM) return;
    const int c  = idx % DIM;
    int t        = idx / DIM;
    const int wn = t % WO; t /= WO;
    const int hn = t % HO;
    const int b  = t / HO;
    const float* base = x + ((size_t)((b * NH + 2 * hn) * NW + 2 * wn)) * DIM + c;
    const float s = base[0] + base[DIM] + base[(size_t)NW * DIM] + base[(size_t)NW * DIM + DIM];
    pooled[idx] = 0.25f * s;
}

// =====================================================================
// Scaled softmax over the last q-dimension (9 elements), in place.
// a layout: (NPOOL, ACOLS) with channel = h*81 + p*9 + q
// =====================================================================
__global__ __launch_bounds__(256)
void softmax_kernel(float* __restrict__ a)
{
    const int row = blockIdx.x * 256 + threadIdx.x;   // NPOOL*HEADS*KK rows
    if (row >= NPOOL * HEADS * KK) return;
    const int p = row % KK;
    int t       = row / KK;
    const int h = t % HEADS;
    const int n = t / HEADS;
    float* r = a + (size_t)n * ACOLS + h * (KK * KK) + p * KK;

    float v[KK];
    float mx = -INFINITY;
#pragma unroll
    for (int q = 0; q < KK; ++q) { v[q] = r[q] * ASCALE; mx = fmaxf(mx, v[q]); }
    float s = 0.0f;
#pragma unroll
    for (int q = 0; q < KK; ++q) { v[q] = __expf(v[q] - mx); s += v[q]; }
    const float inv = 1.0f / s;
#pragma unroll
    for (int q = 0; q < KK; ++q) r[q] = v[q] * inv;
}

// =====================================================================
// Attention apply + fold, gather formulation (no atomics, deterministic):
//   y[b,i,j,d] = sum over valid (pi,pj) windows (hn=(i+1-pi)/2, wn=(j+1-pj)/2)
//                sum_q a[b,n,h,(pi,pj),q] * v[b, 2hn+qi-1, 2wn+qj-1, d]
// block = 192 threads = one pixel; d = threadIdx.x; head h = d>>5.
// =====================================================================
__global__ __launch_bounds__(192)
void attn_fold_kernel(const float* __restrict__ a, const float* __restrict__ v,
                      float* __restrict__ y)
{
    const int pix = blockIdx.x;            // 0..NPIX-1
    const int d   = threadIdx.x;           // 0..191
    const int j   = pix % NW;
    int t         = pix / NW;
    const int i   = t % NH;
    const int b   = t / NH;
    const int h   = d >> 5;                // head index

    float sum = 0.0f;
#pragma unroll
    for (int pi = 0; pi < KWIN; ++pi) {
        const int ni = i + 1 - pi;
        if (ni & 1) continue;              // parity: must be even for stride 2
        const int hn = ni >> 1;
        if (hn < 0 || hn >= HO) continue;
#pragma unroll
        for (int pj = 0; pj < KWIN; ++pj) {
            const int nj = j + 1 - pj;
            if (nj & 1) continue;
            const int wn = nj >> 1;
            if (wn < 0 || wn >= WO) continue;
            const float* arow = a + (size_t)((b * HO + hn) * WO + wn) * ACOLS
                                  + h * (KK * KK) + (pi * KWIN + pj) * KK;
#pragma unroll
            for (int qi = 0; qi < KWIN; ++qi) {
                const int vi = 2 * hn + qi - 1;
                const bool iok = (vi >= 0) && (vi < NH);
#pragma unroll
                for (int qj = 0; qj < KWIN; ++qj) {
                    const int vj = 2 * wn + qj - 1;
                    float vv = 0.0f;
                    if (iok && vj >= 0 && vj < NW)
                        vv = v[(size_t)((b * NH + vi) * NW + vj) * DIM + d];
                    sum = fmaf(arow[qi * KWIN + qj], vv, sum);
                }
            }
        }
    }
    y[(size_t)pix * DIM + d] = sum;
}

// =====================================================================
// launch
// =====================================================================
extern "C" void kernel_launch(void* const* d_in, const int* in_sizes, int n_in,
                              void* d_out, int out_size, void* d_ws, size_t ws_size,
                              hipStream_t stream)
{
    (void)in_sizes; (void)n_in; (void)out_size; (void)ws_size;
    const float* x  = (const float*)d_in[0];
    const float* Wv = (const float*)d_in[1];
    const float* Wa = (const float*)d_in[2];
    const float* ba = (const float*)d_in[3];
    const float* Wp = (const float*)d_in[4];
    const float* bp = (const float*)d_in[5];
    float* out = (float*)d_out;

    // workspace carving (~111 MB total)
    float* v      = (float*)d_ws;                        // NPIX  * DIM
    float* pooled = v      + (size_t)NPIX  * DIM;        // NPOOL * DIM
    float* a      = pooled + (size_t)NPOOL * DIM;        // NPOOL * ACOLS
    float* y      = a      + (size_t)NPOOL * ACOLS;      // NPIX  * DIM

    constexpr int NT   = 4;        // 64 output columns per block
    constexpr int NCOL = 16 * NT;

    // 1) v = x @ Wv                      (M=50176, N=192, K=192)
    gemm_wmma_nt<NT><<<dim3(NPIX / 64, (DIM + NCOL - 1) / NCOL), 128, 0, stream>>>(
        x, Wv, nullptr, v, NPIX, DIM, DIM);

    // 2) pooled = avgpool2x2(x)
    {
        const int n = NPOOL * DIM;
        pool_kernel<<<(n + 255) / 256, 256, 0, stream>>>(x, pooled);
    }

    // 3) a = pooled @ Wa + ba            (M=12544, N=486, K=192)
    gemm_wmma_nt<NT><<<dim3(NPOOL / 64, (ACOLS + NCOL - 1) / NCOL), 128, 0, stream>>>(
        pooled, Wa, ba, a, NPOOL, ACOLS, DIM);

    // 4) scaled softmax over q (in place)
    {
        const int n = NPOOL * HEADS * KK;
        softmax_kernel<<<(n + 255) / 256, 256, 0, stream>>>(a);
    }

    // 5) y = fold(attention-apply(a, patches(v)))
    attn_fold_kernel<<<NPIX, DIM, 0, stream>>>(a, v, y);

    // 6) out = y @ Wp + bp               (M=50176, N=192, K=192)
    gemm_wmma_nt<NT><<<dim3(NPIX / 64, (DIM + NCOL - 1) / NCOL), 128, 0, stream>>>(
        y, Wp, bp, out, NPIX, DIM, DIM);
}